// Aggregator_57878979281442
// MI455X (gfx1250) — compile-verified
//
#include <hip/hip_runtime.h>

#define NG     100000
#define NS     20000
#define EDGES  300000
#define DIM    256
#define NQ     4096
#define MAXU   4096        // max unique queried nodes per side (<= NQ)
#define LDSPAD 2           // row padding: stride 258 floats -> bank-conflict free

typedef __attribute__((ext_vector_type(2))) float v2f;
typedef __attribute__((ext_vector_type(8))) float v8f;

// ---------------------------------------------------------------- init slots
__global__ __launch_bounds__(256) void k_init_slots(int* __restrict__ sg,
                                                    int* __restrict__ ss,
                                                    int* __restrict__ cnt) {
  int i = blockIdx.x * 256 + threadIdx.x;
  if (i < NG) sg[i] = -1;
  if (i < NS) ss[i] = -1;
  if (i < 2)  cnt[i] = 0;
}

// ---------------------------------------------------------------- zero nh
__global__ __launch_bounds__(256) void k_zero(float* __restrict__ p, int n) {
  int i = blockIdx.x * 256 + threadIdx.x;
  if (i < n) p[i] = 0.0f;
}

// ------------------------------------------------- mark queried nodes, compact
__global__ __launch_bounds__(256) void k_compact(const int* __restrict__ gid,
                                                 const int* __restrict__ sid,
                                                 int* __restrict__ sg,
                                                 int* __restrict__ ss,
                                                 int* __restrict__ cnt) {
  int i = blockIdx.x * 256 + threadIdx.x;
  if (i < NQ) {
    int g = gid[i];
    if (atomicCAS(&sg[g], -1, -2) == -1) sg[g] = atomicAdd(&cnt[0], 1);
  } else if (i < 2 * NQ) {
    int g = sid[i - NQ];
    if (atomicCAS(&ss[g], -1, -2) == -1) ss[g] = atomicAdd(&cnt[1], 1);
  }
}

// ---------------------------------- filtered edge scatter: one wave per edge
__global__ __launch_bounds__(256) void k_scatter(const float* __restrict__ vsrc,
                                                 const float* __restrict__ att,
                                                 const int* __restrict__ src,
                                                 const int* __restrict__ dst,
                                                 const int* __restrict__ slot,
                                                 float* __restrict__ nh, int nE) {
  const int wid  = (blockIdx.x * 256 + threadIdx.x) >> 5;   // edge id (wave32)
  const int lane = threadIdx.x & 31;
  if (wid >= nE) return;
  const int s = slot[dst[wid]];
  if (s < 0) return;                 // destination never queried: skip edge
  const float a = att[wid];
  const float4* vp = (const float4*)(vsrc + (size_t)src[wid] * DIM);
  float* np = nh + (size_t)s * DIM;
#pragma unroll
  for (int j = 0; j < 2; ++j) {
    const int q = lane + j * 32;     // float4 index 0..63 over 256 floats
    const float4 vv = vp[q];
    atomicAdd(np + q * 4 + 0, vv.x * a);
    atomicAdd(np + q * 4 + 1, vv.y * a);
    atomicAdd(np + q * 4 + 2, vv.z * a);
    atomicAdd(np + q * 4 + 3, vv.w * a);
  }
}

// ------------- combine: per 16 query rows, two GEMV-batches on WMMA f32 cores
// out[m][n] = lrelu((v+nh) @ W1^T + b1)[m][n] + lrelu((v*nh) @ W1^T + b1)[m][n]
__global__ __launch_bounds__(32) void k_combine(const float* __restrict__ v,
                                                const float* __restrict__ nh,
                                                const int* __restrict__ slot,
                                                const int* __restrict__ ids,
                                                const float* __restrict__ W1,
                                                const float* __restrict__ b1,
                                                float* __restrict__ out) {
  __shared__ float x1[16][DIM + LDSPAD];
  __shared__ float x2[16][DIM + LDSPAD];
  __shared__ int   qg[16];
  __shared__ int   qs[16];

  const int lane  = threadIdx.x;         // 0..31, one wave per block
  const int qbase = blockIdx.x * 16;

  if (lane < 16) {
    const int g = ids[qbase + lane];
    qg[lane] = g;
    qs[lane] = slot[g];                  // guaranteed >= 0 (marked in compact)
  }
  __syncthreads();

  // stage x1 = v + nh, x2 = v * nh into LDS (16 rows x 256, float4 traffic)
  for (int i = lane; i < 16 * (DIM / 4); i += 32) {
    const int r = i >> 6;                // DIM/4 = 64 float4 per row
    const int c = (i & 63) * 4;
    const float4 vv = *(const float4*)(v  + (size_t)qg[r] * DIM + c);
    const float4 nn = *(const float4*)(nh + (size_t)qs[r] * DIM + c);
    x1[r][c + 0] = vv.x + nn.x;  x2[r][c + 0] = vv.x * nn.x;
    x1[r][c + 1] = vv.y + nn.y;  x2[r][c + 1] = vv.y * nn.y;
    x1[r][c + 2] = vv.z + nn.z;  x2[r][c + 2] = vv.z * nn.z;
    x1[r][c + 3] = vv.w + nn.w;  x2[r][c + 3] = vv.w * nn.w;
  }
  __syncthreads();

  // WMMA f32 16x16x4 layouts (ISA 7.12.2):
  //   A (16x4): lane = {khalf[4], mrow[3:0]}, VGPR0/1 = K = 2*khalf + {0,1}
  //   B (4x16): lane = {khalf[4], ncol[3:0]}, VGPR0/1 = K = 2*khalf + {0,1}
  //   C/D (16x16): VGPR r, row = khalf*8 + r, col = lane&15
  const int mrow  = lane & 15;
  const int khalf = lane >> 4;

  for (int n0 = 0; n0 < DIM; n0 += 16) {
    v8f acc1 = {};
    v8f acc2 = {};
    const int ncol = n0 + (lane & 15);
    const float* wrow = W1 + (size_t)ncol * DIM;   // W1[ncol][k], row-major

#pragma unroll 4
    for (int k0 = 0; k0 < DIM; k0 += 4) {
      const int ka = k0 + 2 * khalf;
      v2f a1; a1.x = x1[mrow][ka]; a1.y = x1[mrow][ka + 1];
      v2f a2; a2.x = x2[mrow][ka]; a2.y = x2[mrow][ka + 1];
      v2f bb; bb.x = wrow[ka];     bb.y = wrow[ka + 1];
      acc1 = __builtin_amdgcn_wmma_f32_16x16x4_f32(
          false, a1, false, bb, (short)0, acc1, false, false);
      acc2 = __builtin_amdgcn_wmma_f32_16x16x4_f32(
          false, a2, false, bb, (short)0, acc2, false, false);
    }

    const float bias = b1[ncol];
#pragma unroll
    for (int r = 0; r < 8; ++r) {
      const int m = khalf * 8 + r;
      float y1 = acc1[r] + bias;
      float y2 = acc2[r] + bias;
      y1 = (y1 >= 0.0f) ? y1 : 0.01f * y1;
      y2 = (y2 >= 0.0f) ? y2 : 0.01f * y2;
      out[(size_t)(qbase + m) * DIM + ncol] = y1 + y2;
    }
  }
}

// ---------------------------------------------------------------------------
extern "C" void kernel_launch(void* const* d_in, const int* in_sizes, int n_in,
                              void* d_out, int out_size, void* d_ws, size_t ws_size,
                              hipStream_t stream) {
  const float* v_grid  = (const float*)d_in[0];
  const float* v_sc    = (const float*)d_in[1];
  const float* att_s2g = (const float*)d_in[2];
  const float* att_g2s = (const float*)d_in[3];
  const float* W1      = (const float*)d_in[4];
  const float* b1      = (const float*)d_in[5];
  const int*   src_s2g = (const int*)d_in[6];
  const int*   dst_s2g = (const int*)d_in[7];
  const int*   src_g2s = (const int*)d_in[8];
  const int*   dst_g2s = (const int*)d_in[9];
  const int*   sc_id   = (const int*)d_in[10];
  const int*   gr_id   = (const int*)d_in[11];
  float* out = (float*)d_out;

  // workspace layout: slot maps + counters + compact nh buffers (~8.9 MB)
  int*   sg  = (int*)d_ws;                         // NG
  int*   ss  = sg + NG;                            // NS
  int*   cnt = ss + NS;                            // 4 (2 used)
  float* nhg = (float*)(cnt + 4);                  // MAXU*DIM
  float* nhs = nhg + (size_t)MAXU * DIM;           // MAXU*DIM

  k_init_slots<<<(NG + 255) / 256, 256, 0, stream>>>(sg, ss, cnt);
  k_zero<<<(2 * MAXU * DIM + 255) / 256, 256, 0, stream>>>(nhg, 2 * MAXU * DIM);
  k_compact<<<(2 * NQ + 255) / 256, 256, 0, stream>>>(gr_id, sc_id, sg, ss, cnt);

  // sc -> grid edges feed nh_grid; grid -> sc edges feed nh_sc
  k_scatter<<<(EDGES + 7) / 8, 256, 0, stream>>>(v_sc,   att_s2g, src_s2g, dst_s2g, sg, nhg, EDGES);
  k_scatter<<<(EDGES + 7) / 8, 256, 0, stream>>>(v_grid, att_g2s, src_g2s, dst_g2s, ss, nhs, EDGES);

  // output order: sc_out[small_category_id] first, then grid_out[grid_id]
  k_combine<<<NQ / 16, 32, 0, stream>>>(v_sc,   nhs, ss, sc_id, W1, b1, out);
  k_combine<<<NQ / 16, 32, 0, stream>>>(v_grid, nhg, sg, gr_id, W1, b1, out + (size_t)NQ * DIM);
}